// UnstructuredSparse_23381801960228
// MI455X (gfx1250) — compile-verified
//
#include <hip/hip_runtime.h>
#include <hip/hip_bf16.h>

// Problem constants (match reference)
#define N_OUT   4096
#define N_IN    4096
#define TOKENS  256

typedef __attribute__((ext_vector_type(2))) float v2f;
typedef __attribute__((ext_vector_type(8))) float v8f;

// ---------------------------------------------------------------------------
// Kernel 1: zero the dense weight buffer W (16.7M floats) with float4 stores
// ---------------------------------------------------------------------------
__global__ void us_zero_w(float4* __restrict__ w4) {
    const int i = blockIdx.x * blockDim.x + threadIdx.x;   // one float4 each
    w4[i] = make_float4(0.f, 0.f, 0.f, 0.f);
}

// ---------------------------------------------------------------------------
// Kernel 2: scatter sparse values into W: W[flat_idx[i]] = val[i]
// ---------------------------------------------------------------------------
__global__ void us_scatter_w(float* __restrict__ W,
                             const float* __restrict__ val,
                             const int* __restrict__ idx,
                             int n) {
    const int i = blockIdx.x * blockDim.x + threadIdx.x;
    if (i < n) {
        W[idx[i]] = val[i];
    }
}

// ---------------------------------------------------------------------------
// Kernel 3: out[t,o] = sum_i x[t,i] * W[o,i]   via V_WMMA_F32_16X16X4_F32
//
// Register-blocked 2x2: each wave computes a 32(M) x 32(N) output tile with
// four v8f accumulators. Per K-step of 4 we load 2 A float2's + 2 B float2's
// and issue 4 WMMAs -> 1 b64 load per WMMA (vs 2 for a 16x16 tile), halving
// L2 load traffic. Block = 8 waves arranged 2(M) x 4(N) -> 64M x 128N tile.
// Grid = (N_OUT/128, TOKENS/64) = (32, 4).
//
// f32 WMMA operand layout (ISA 7.12.2, 32-bit A 16x4):
//   lanes 0-15  : row = lane,    holds K = {0,1} in v[0],v[1]
//   lanes 16-31 : row = lane-16, holds K = {2,3}
// Both A (M x K) and B rows (W is [o, i], K-contiguous) are contiguous
// float2 loads per lane. C/D layout: VGPR v -> M=v (lanes 0-15), M=v+8
// (lanes 16-31); N = lane%16.
// ---------------------------------------------------------------------------
__global__ void __launch_bounds__(256)
us_gemm_wmma(const float* __restrict__ x,    // [TOKENS, N_IN]
             const float* __restrict__ W,    // [N_OUT, N_IN]
             float* __restrict__ out) {      // [TOKENS, N_OUT]
    const int lane  = threadIdx.x & 31;
    const int wave  = threadIdx.x >> 5;
    const int wm    = wave & 1;               // 2 waves along M
    const int wn    = wave >> 1;              // 4 waves along N

    const int m0 = blockIdx.y * 64 + wm * 32;       // token tile base (2 tiles)
    const int n0 = blockIdx.x * 128 + wn * 32;      // out-channel tile base (2)

    const int r     = lane & 15;          // row-within-16-tile carried by lane
    const int khalf = (lane >> 4) * 2;    // 0 for lanes 0-15, 2 for lanes 16-31

    const float* a0p = x + (size_t)(m0 + r) * N_IN + khalf;
    const float* a1p = a0p + (size_t)16 * N_IN;
    const float* b0p = W + (size_t)(n0 + r) * N_IN + khalf;
    const float* b1p = b0p + (size_t)16 * N_IN;

    v8f acc00 = {};   // (m-tile 0, n-tile 0)
    v8f acc01 = {};   // (m-tile 0, n-tile 1)
    v8f acc10 = {};   // (m-tile 1, n-tile 0)
    v8f acc11 = {};   // (m-tile 1, n-tile 1)

    // K loop: 4 per WMMA group, unrolled x2 (8 K per iteration).
    for (int k = 0; k < N_IN; k += 8) {
        // Prefetch upcoming W panels (~1KB ahead) for both B row groups.
        __builtin_prefetch(b0p + k + 256, 0, 1);
        __builtin_prefetch(b1p + k + 256, 0, 1);

        v2f a0 = *(const v2f*)(a0p + k);
        v2f a1 = *(const v2f*)(a1p + k);
        v2f b0 = *(const v2f*)(b0p + k);
        v2f b1 = *(const v2f*)(b1p + k);
        v2f a2 = *(const v2f*)(a0p + k + 4);
        v2f a3 = *(const v2f*)(a1p + k + 4);
        v2f b2 = *(const v2f*)(b0p + k + 4);
        v2f b3 = *(const v2f*)(b1p + k + 4);

        acc00 = __builtin_amdgcn_wmma_f32_16x16x4_f32(
                    false, a0, false, b0, (short)0, acc00, false, false);
        acc01 = __builtin_amdgcn_wmma_f32_16x16x4_f32(
                    false, a0, false, b1, (short)0, acc01, false, false);
        acc10 = __builtin_amdgcn_wmma_f32_16x16x4_f32(
                    false, a1, false, b0, (short)0, acc10, false, false);
        acc11 = __builtin_amdgcn_wmma_f32_16x16x4_f32(
                    false, a1, false, b1, (short)0, acc11, false, false);

        acc00 = __builtin_amdgcn_wmma_f32_16x16x4_f32(
                    false, a2, false, b2, (short)0, acc00, false, false);
        acc01 = __builtin_amdgcn_wmma_f32_16x16x4_f32(
                    false, a2, false, b3, (short)0, acc01, false, false);
        acc10 = __builtin_amdgcn_wmma_f32_16x16x4_f32(
                    false, a3, false, b2, (short)0, acc10, false, false);
        acc11 = __builtin_amdgcn_wmma_f32_16x16x4_f32(
                    false, a3, false, b3, (short)0, acc11, false, false);
    }

    // Store D tiles: VGPR v -> M = v + (lane<16 ? 0 : 8), N = lane%16.
    const int mbase = m0 + ((lane < 16) ? 0 : 8);
    {
        float* o00 = out + (size_t)mbase * N_OUT + (n0 + r);
        float* o01 = o00 + 16;                         // n-tile 1
        float* o10 = o00 + (size_t)16 * N_OUT;         // m-tile 1
        float* o11 = o10 + 16;
#pragma unroll
        for (int v = 0; v < 8; ++v) {
            const size_t off = (size_t)v * N_OUT;
            o00[off] = acc00[v];
            o01[off] = acc01[v];
            o10[off] = acc10[v];
            o11[off] = acc11[v];
        }
    }
}

// ---------------------------------------------------------------------------
// Launch: zero W (d_ws) -> scatter -> WMMA GEMM. All on `stream`, graph-safe.
// d_ws must hold the dense W: 4096*4096*4 = 64 MB.
// ---------------------------------------------------------------------------
extern "C" void kernel_launch(void* const* d_in, const int* in_sizes, int n_in,
                              void* d_out, int out_size, void* d_ws, size_t ws_size,
                              hipStream_t stream) {
    const float* x   = (const float*)d_in[0];   // [256, 4096] f32
    const float* val = (const float*)d_in[1];   // [n_sparse]  f32
    const int*   idx = (const int*)d_in[2];     // [n_sparse]  i32
    const int n_sparse = in_sizes[1];

    float* W = (float*)d_ws;                    // dense [N_OUT, N_IN]
    float* out = (float*)d_out;                 // [TOKENS, N_OUT]

    // 1) zero W: 16777216 floats = 4194304 float4
    {
        const int threads = 256;
        const int total4  = (N_OUT * N_IN) / 4;
        us_zero_w<<<total4 / threads, threads, 0, stream>>>((float4*)W);
    }
    // 2) scatter nonzeros
    {
        const int threads = 256;
        const int blocks  = (n_sparse + threads - 1) / threads;
        us_scatter_w<<<blocks, threads, 0, stream>>>(W, val, idx, n_sparse);
    }
    // 3) GEMM via f32 WMMA, 2x2 register-blocked
    {
        dim3 grid(N_OUT / 128, TOKENS / 64);
        us_gemm_wmma<<<grid, 256, 0, stream>>>(x, W, out);
    }
}